// CMUNeXtBlock_MK_resiual2_29850022707799
// MI455X (gfx1250) — compile-verified
//
#include <hip/hip_runtime.h>
#include <hip/hip_bf16.h>

typedef __attribute__((ext_vector_type(2))) float v2f;
typedef __attribute__((ext_vector_type(8))) float v8f;
typedef int v4i __attribute__((vector_size(16)));

#define Sp   262144      // 64*64*64
#define HWc  4096
#define Wc   64

#if defined(__HIP_DEVICE_COMPILE__) && __has_builtin(__builtin_amdgcn_global_load_async_to_lds_b128)
#define HAVE_ASYNC_LDS 1
typedef __attribute__((address_space(1))) v4i* gv4i_p;
typedef __attribute__((address_space(3))) v4i* sv4i_p;
#else
#define HAVE_ASYNC_LDS 0
#endif

__device__ __forceinline__ void wait_asynccnt0() {
#if defined(__HIP_DEVICE_COMPILE__)
#if __has_builtin(__builtin_amdgcn_s_wait_asynccnt)
  __builtin_amdgcn_s_wait_asynccnt(0);
#else
  asm volatile("s_wait_asynccnt 0x0" ::: "memory");
#endif
#endif
}

// ---------------- stats zero ----------------
__global__ void k_zero(float* s) { s[threadIdx.x] = 0.0f; }

// ---------------- stage 1: depthwise convs + residual, GN partial sums ----
__global__ void k_dw(const float* __restrict__ x,
                     const float* __restrict__ w3, const float* __restrict__ b3,
                     const float* __restrict__ w5, const float* __restrict__ b5,
                     const float* __restrict__ w7, const float* __restrict__ b7,
                     const float* __restrict__ w9, const float* __restrict__ b9,
                     float* __restrict__ y,
                     float* __restrict__ gs, float* __restrict__ gq) {
  int i = blockIdx.x * 256 + threadIdx.x;
  int b = i / (32 * Sp);
  int rem = i - b * 32 * Sp;
  int c = rem / Sp;
  int s = rem - c * Sp;
  int d = s >> 12, h = (s >> 6) & 63, w = s & 63;
  int g = c >> 3, cg = c & 7;
  int k = 3 + 2 * g, pd = g + 1;
  const float *wt, *bs;
  if (g == 0)      { wt = w3; bs = b3; }
  else if (g == 1) { wt = w5; bs = b5; }
  else if (g == 2) { wt = w7; bs = b7; }
  else             { wt = w9; bs = b9; }
  wt += cg * k * k * k;
  const float* xc = x + (b * 32 + c) * Sp;
  float acc = bs[cg];
  for (int kd = 0; kd < k; ++kd) {
    int zd = d + kd - pd;
    if (zd < 0 || zd >= 64) continue;
    int zdn = zd + 1;
    if (zdn < 64) __builtin_prefetch(xc + zdn * HWc + h * Wc, 0, 1);
    for (int kh = 0; kh < k; ++kh) {
      int zh = h + kh - pd;
      if (zh < 0 || zh >= 64) continue;
      const float* row  = xc + zd * HWc + zh * Wc;
      const float* wrow = wt + (kd * k + kh) * k;
      for (int kw = 0; kw < k; ++kw) {
        int zw = w + kw - pd;
        if (zw < 0 || zw >= 64) continue;
        acc += row[zw] * wrow[kw];
      }
    }
  }
  float v = acc + x[i];
  y[i] = v;
  // block covers one (b,c): reduce sum/sumsq for GroupNorm
  __shared__ float rs_[256], rq_[256];
  rs_[threadIdx.x] = v; rq_[threadIdx.x] = v * v;
  __syncthreads();
  for (int off = 128; off > 0; off >>= 1) {
    if (threadIdx.x < off) {
      rs_[threadIdx.x] += rs_[threadIdx.x + off];
      rq_[threadIdx.x] += rq_[threadIdx.x + off];
    }
    __syncthreads();
  }
  if (threadIdx.x == 0) {
    atomicAdd(&gs[b * 4 + g], rs_[0]);
    atomicAdd(&gq[b * 4 + g], rq_[0]);
  }
}

// ---------------- finalize mean / rstd ----------------
__global__ void k_fin(const float* __restrict__ sum, const float* __restrict__ sqs,
                      float* __restrict__ mu, float* __restrict__ rs,
                      int n, float invcnt) {
  int i = threadIdx.x;
  if (i < n) {
    float m = sum[i] * invcnt;
    float v = sqs[i] * invcnt - m * m;
    mu[i] = m;
    rs[i] = rsqrtf(v + 1e-5f);
  }
}

// ---------------- stage 2: GN+GELU fused into WMMA pointwise 32->8 -------
__global__ void k_gn_gelu_pw(const float* __restrict__ y,
                             const float* __restrict__ gmu, const float* __restrict__ grs,
                             const float* __restrict__ gnw, const float* __restrict__ gnb,
                             const float* __restrict__ w_pw,
                             float* __restrict__ p,
                             float* __restrict__ s1, float* __restrict__ q1) {
  __shared__ float zs[32 * 128];        // [c][m] post GN+GELU A-tile
  int b  = blockIdx.x >> 11;            // / 2048 tiles per batch
  int s0 = (blockIdx.x & 2047) * 128;
  int tid = threadIdx.x;
  for (int j = 0; j < 16; ++j) {
    int idx = tid + j * 256;
    int c = idx >> 7, m = idx & 127, g = c >> 3;
    float v = y[(b * 32 + c) * Sp + s0 + m];
    v = (v - gmu[b * 4 + g]) * grs[b * 4 + g] * gnw[c] + gnb[c];
    v = 0.5f * v * (1.0f + erff(v * 0.70710678118654752f));   // exact GELU
    zs[c * 128 + m] = v;
  }
  __syncthreads();
  int wv = tid >> 5, lane = tid & 31;
  int m0 = wv * 16;
  int l15 = lane & 15;
  int hi = (lane >> 4) & 1;            // lane half selects K-pair / M-half
  float msk = (l15 < 8) ? 1.0f : 0.0f; // only 8 real output channels
  int o = l15 & 7;
  v8f acc = {};
  #pragma unroll
  for (int kb = 0; kb < 8; ++kb) {     // K = 32 input channels, 4 per step
    int ka = kb * 4 + 2 * hi;
    v2f A, Bv;
    A.x = zs[(ka + 0) * 128 + m0 + l15];
    A.y = zs[(ka + 1) * 128 + m0 + l15];
    Bv.x = w_pw[o * 32 + ka + 0] * msk;
    Bv.y = w_pw[o * 32 + ka + 1] * msk;
    acc = __builtin_amdgcn_wmma_f32_16x16x4_f32(false, A, false, Bv,
                                                (short)0, acc, false, false);
  }
  if (l15 < 8) {                       // D layout: 8 contiguous M per lane
    float* base = p + (b * 8 + l15) * Sp + s0 + m0 + hi * 8;
    float ls = 0.0f, lq = 0.0f;
    #pragma unroll
    for (int i = 0; i < 8; ++i) { float v = acc[i]; base[i] = v; ls += v; lq += v * v; }
    atomicAdd(&s1[b * 8 + l15], ls);
    atomicAdd(&q1[b * 8 + l15], lq);
  }
}

// ---------------- stage 3a: IN1 + SiLU elementwise -> q ----------------
__global__ void k_act(const float* __restrict__ p,
                      const float* __restrict__ mu1, const float* __restrict__ rs1,
                      float* __restrict__ q) {
  int i = blockIdx.x * 256 + threadIdx.x;
  int bc = i / Sp;                     // b*8+c in 0..15
  float t = (p[i] - mu1[bc]) * rs1[bc];
  q[i] = t / (1.0f + expf(-t));
}

// ---------------- stage 3b: WMMA 3x3x3 conv 8->32 (reads activated q) ----
__global__ void k_conv3(const float* __restrict__ q,
                        const float* __restrict__ w_nxn,
                        float* __restrict__ r,
                        float* __restrict__ s2, float* __restrict__ q2) {
  __shared__ float qt[8 * 9 * 66];     // [ci][td*3+th][w+1]  zero-padded halos
  __shared__ float wl[32 * 8 * 27];    // raw w_nxn layout: [o][ci][tap]
  int b  = blockIdx.x >> 12;           // / (64*64)
  int dh = blockIdx.x & 4095;
  int d = dh >> 6, h = dh & 63;
  int tid = threadIdx.x;

  // weights: pure aligned copy of 27648 B -> async global->LDS (B128) path
#if HAVE_ASYNC_LDS
  for (int idx = tid; idx < 1728; idx += 256) {   // 1728 x 16B chunks
    __builtin_amdgcn_global_load_async_to_lds_b128(
        (gv4i_p)(w_nxn + idx * 4),
        (sv4i_p)(wl + idx * 4),
        0, 0);
  }
#else
  for (int idx = tid; idx < 32 * 8 * 27; idx += 256) wl[idx] = w_nxn[idx];
#endif

  // activated-input tile with zero halos
  for (int idx = tid; idx < 8 * 9 * 66; idx += 256) {
    int ci = idx / 594;
    int rem = idx - ci * 594;
    int tdh = rem / 66;
    int ww = rem - tdh * 66;
    int zd = d + tdh / 3 - 1, zh = h + tdh % 3 - 1, zw = ww - 1;
    float v = 0.0f;
    if (zd >= 0 && zd < 64 && zh >= 0 && zh < 64 && zw >= 0 && zw < 64)
      v = q[(b * 8 + ci) * Sp + zd * HWc + zh * Wc + zw];
    qt[idx] = v;
  }
#if HAVE_ASYNC_LDS
  wait_asynccnt0();
#endif
  __syncthreads();

  int wv = tid >> 5, lane = tid & 31;
  int mt = wv >> 1, nt = wv & 1;       // 4 M-tiles x 2 N-tiles
  int w0 = mt * 16, o0 = nt * 16;
  int l15 = lane & 15, hi = (lane >> 4) & 1;
  int ob = (o0 + l15) * 216;           // this lane's output-channel weight row
  v8f acc = {};
  for (int tap = 0; tap < 27; ++tap) { // 27 shifted K=8 GEMMs
    int tdh = (tap / 9) * 3 + (tap / 3) % 3;
    int tw  = tap % 3;
    #pragma unroll
    for (int kb = 0; kb < 2; ++kb) {
      int ka = kb * 4 + 2 * hi;
      v2f A, Bv;
      A.x = qt[(ka + 0) * 594 + tdh * 66 + w0 + l15 + tw];
      A.y = qt[(ka + 1) * 594 + tdh * 66 + w0 + l15 + tw];
      Bv.x = wl[ob + (ka + 0) * 27 + tap];
      Bv.y = wl[ob + (ka + 1) * 27 + tap];
      acc = __builtin_amdgcn_wmma_f32_16x16x4_f32(false, A, false, Bv,
                                                  (short)0, acc, false, false);
    }
  }
  int o = o0 + l15;
  float* base = r + (b * 32 + o) * Sp + d * HWc + h * Wc + w0 + hi * 8;
  float ls = 0.0f, lq = 0.0f;
  #pragma unroll
  for (int i = 0; i < 8; ++i) { float v = acc[i]; base[i] = v; ls += v; lq += v * v; }
  atomicAdd(&s2[b * 32 + o], ls);
  atomicAdd(&q2[b * 32 + o], lq);
}

// ---------------- stage 4: IN2 + SiLU + residual ----------------
__global__ void k_out(const float* __restrict__ r,
                      const float* __restrict__ mu2, const float* __restrict__ rs2,
                      const float* __restrict__ x, float* __restrict__ out) {
  int i = blockIdx.x * 256 + threadIdx.x;
  int bc = i / Sp;
  float t = (r[i] - mu2[bc]) * rs2[bc];
  t = t / (1.0f + expf(-t));
  out[i] = t + x[i];
}

extern "C" void kernel_launch(void* const* d_in, const int* in_sizes, int n_in,
                              void* d_out, int out_size, void* d_ws, size_t ws_size,
                              hipStream_t stream) {
  (void)in_sizes; (void)n_in; (void)out_size; (void)ws_size;
  const float* x    = (const float*)d_in[0];
  const float* w3   = (const float*)d_in[1];
  const float* b3   = (const float*)d_in[2];
  const float* w5   = (const float*)d_in[3];
  const float* b5   = (const float*)d_in[4];
  const float* w7   = (const float*)d_in[5];
  const float* b7   = (const float*)d_in[6];
  const float* w9   = (const float*)d_in[7];
  const float* b9   = (const float*)d_in[8];
  const float* gnw  = (const float*)d_in[9];
  const float* gnb  = (const float*)d_in[10];
  const float* w_pw = (const float*)d_in[11];
  const float* w_nxn= (const float*)d_in[12];
  float* out = (float*)d_out;

  const int N = 2 * 32 * Sp;           // 16,777,216
  const int Np = 2 * 8 * Sp;           // 4,194,304
  float* ws    = (float*)d_ws;
  float* stats = ws;                   // 512 floats reserved
  float* y     = ws + 512;             // N
  float* p     = y + N;                // Np
  float* r     = p + Np;               // N
  float* q     = r + N;                // Np
  // stats layout
  float* gn_sum = stats;        float* gn_sqs = stats + 8;
  float* gn_mu  = stats + 16;   float* gn_rs  = stats + 24;
  float* i1_sum = stats + 32;   float* i1_sqs = stats + 48;
  float* i1_mu  = stats + 64;   float* i1_rs  = stats + 80;
  float* i2_sum = stats + 96;   float* i2_sqs = stats + 160;
  float* i2_mu  = stats + 224;  float* i2_rs  = stats + 288;

  k_zero<<<1, 512, 0, stream>>>(stats);

  k_dw<<<N / 256, 256, 0, stream>>>(x, w3, b3, w5, b5, w7, b7, w9, b9,
                                    y, gn_sum, gn_sqs);
  k_fin<<<1, 32, 0, stream>>>(gn_sum, gn_sqs, gn_mu, gn_rs, 8,
                              1.0f / (8.0f * (float)Sp));

  k_gn_gelu_pw<<<2 * (Sp / 128), 256, 0, stream>>>(y, gn_mu, gn_rs, gnw, gnb,
                                                   w_pw, p, i1_sum, i1_sqs);
  k_fin<<<1, 32, 0, stream>>>(i1_sum, i1_sqs, i1_mu, i1_rs, 16,
                              1.0f / (float)Sp);

  k_act<<<Np / 256, 256, 0, stream>>>(p, i1_mu, i1_rs, q);

  k_conv3<<<2 * 64 * 64, 256, 0, stream>>>(q, w_nxn, r, i2_sum, i2_sqs);
  k_fin<<<1, 64, 0, stream>>>(i2_sum, i2_sqs, i2_mu, i2_rs, 64,
                              1.0f / (float)Sp);

  k_out<<<N / 256, 256, 0, stream>>>(r, i2_mu, i2_rs, x, out);
}